// VNInvariantAttention_50019189129715
// MI455X (gfx1250) — compile-verified
//
#include <hip/hip_runtime.h>
#include <math.h>

// Problem constants (from reference setup_inputs)
#define BB 8
#define NN 2048
#define CC 128
#define CO 128
#define HH 32
#define KK 16
#define EPSF 1e-6f
#define CLAMP_MAXF 50.0f

typedef __attribute__((ext_vector_type(2))) float v2f;
typedef __attribute__((ext_vector_type(8))) float v8f;

// ---------------------------------------------------------------------------
// Kernel 1: brute-force KNN (k=16, self excluded), LDS-tiled.
// One thread per query point, register-resident sorted top-16.
// ---------------------------------------------------------------------------
__global__ __launch_bounds__(256) void knn_kernel(const float* __restrict__ x,
                                                  int* __restrict__ idx_out) {
    __shared__ float sx[256], sy[256], sz[256];
    const int t = threadIdx.x;
    const int p = blockIdx.x * 256 + t;     // blocks never straddle batches (2048 % 256 == 0)
    const int b = p / NN;
    const int n = p % NN;
    const float px = x[p * 3 + 0], py = x[p * 3 + 1], pz = x[p * 3 + 2];

    float kd[KK];
    int   ki[KK];
#pragma unroll
    for (int j = 0; j < KK; ++j) { kd[j] = 3.0e38f; ki[j] = 0; }

    for (int m0 = 0; m0 < NN; m0 += 256) {
        const int g = (b * NN + m0 + t) * 3;
        sx[t] = x[g + 0]; sy[t] = x[g + 1]; sz[t] = x[g + 2];
        __syncthreads();
        for (int mm = 0; mm < 256; ++mm) {
            const int m = m0 + mm;
            const float dx = px - sx[mm], dy = py - sy[mm], dz = pz - sz[mm];
            const float d2 = dx * dx + dy * dy + dz * dz;
            if (m != n && d2 < kd[KK - 1]) {
                // fully unrolled bubble-insert keeps arrays in VGPRs
                float cd = d2; int ci = m;
#pragma unroll
                for (int j = 0; j < KK; ++j) {
                    const bool sw = cd < kd[j];
                    const float td = sw ? cd : kd[j];
                    const int   ti = sw ? ci : ki[j];
                    cd = sw ? kd[j] : cd;
                    ci = sw ? ki[j] : ci;
                    kd[j] = td; ki[j] = ti;
                }
            }
        }
        __syncthreads();
    }
#pragma unroll
    for (int j = 0; j < KK; ++j) idx_out[p * KK + j] = ki[j];
}

// ---------------------------------------------------------------------------
// Kernel 2: VN linear  out[o,(p,d)] = sum_c W[o,c] * v[p,c,d]
// GEMM M=128 (split 2 blocks of 64), K=128, N tile = 16 columns.
// fp32 WMMA 16x16x4, 4 waves per block (one 16-row strip each).
// ---------------------------------------------------------------------------
__global__ __launch_bounds__(128) void vn_linear_kernel(
    const float* __restrict__ Wq, const float* __restrict__ Wk, const float* __restrict__ Wu,
    const float* __restrict__ v,
    float* __restrict__ Q, float* __restrict__ Kt, float* __restrict__ U) {
    __shared__ float sW[64][129];   // padded: A-operand reads walk rows
    __shared__ float sV[128][17];   // padded: B-operand reads walk columns

    const int t = threadIdx.x;
    const int col0  = blockIdx.x * 16;
    const int mhalf = blockIdx.y;           // 0/1 -> output rows [0,64)/[64,128)
    const float* W   = (blockIdx.z == 0) ? Wq : ((blockIdx.z == 1) ? Wk : Wu);
    float*       out = (blockIdx.z == 0) ? Q  : ((blockIdx.z == 1) ? Kt : U);

    for (int e = t; e < 64 * 128; e += 128) {
        const int r = e >> 7, c = e & 127;
        sW[r][c] = W[(mhalf * 64 + r) * CC + c];
    }
    for (int e = t; e < 128 * 16; e += 128) {
        const int c = e >> 4, jj = e & 15;
        const int j = col0 + jj;
        const int pp = j / 3, d = j - pp * 3;
        sV[c][jj] = v[pp * (CC * 3) + c * 3 + d];
    }
    __syncthreads();

    const int wave = t >> 5, lane = t & 31;
    const int m0 = wave * 16;
    const int lr = lane & 15;
    const bool hi = lane >= 16;

    v8f acc = {0.f, 0.f, 0.f, 0.f, 0.f, 0.f, 0.f, 0.f};
#pragma unroll 4
    for (int k0 = 0; k0 < CC; k0 += 4) {
        v2f a, bm;
        // A (16x4 fp32): lanes 0-15 hold K={0,1}, lanes 16-31 hold K={2,3}
        a.x  = sW[m0 + lr][k0 + (hi ? 2 : 0)];
        a.y  = sW[m0 + lr][k0 + (hi ? 3 : 1)];
        // B (4x16 fp32): rows striped across lanes within each VGPR
        bm.x = sV[k0 + (hi ? 2 : 0)][lr];
        bm.y = sV[k0 + (hi ? 3 : 1)][lr];
        acc = __builtin_amdgcn_wmma_f32_16x16x4_f32(false, a, false, bm,
                                                    (short)0, acc, false, false);
    }

    // C/D layout: VGPR r -> lanes 0-15: M=m0+r, lanes 16-31: M=m0+8+r, N=lane&15
    const int col = col0 + lr;
    const int pp = col / 3, d = col - pp * 3;
    const int mbase = mhalf * 64 + m0 + (hi ? 8 : 0);
#pragma unroll
    for (int r = 0; r < 8; ++r)
        out[pp * (CO * 3) + (mbase + r) * 3 + d] = acc[r];
}

// ---------------------------------------------------------------------------
// Kernel 3: qn/kn = mean over channels of per-channel 3-vector norm.
// One wave per point (wave32).
// ---------------------------------------------------------------------------
__global__ __launch_bounds__(256) void norms_kernel(const float* __restrict__ Q,
                                                    const float* __restrict__ Kt,
                                                    float* __restrict__ qn,
                                                    float* __restrict__ kn) {
    const int t = threadIdx.x, wave = t >> 5, lane = t & 31;
    const int p = blockIdx.x * 8 + wave;
    float qs = 0.f, ks = 0.f;
#pragma unroll
    for (int i = 0; i < 4; ++i) {
        const int c = lane + i * 32;
        const float* qp = Q  + p * (CO * 3) + c * 3;
        const float* kp = Kt + p * (CO * 3) + c * 3;
        qs += sqrtf(qp[0] * qp[0] + qp[1] * qp[1] + qp[2] * qp[2]);
        ks += sqrtf(kp[0] * kp[0] + kp[1] * kp[1] + kp[2] * kp[2]);
    }
#pragma unroll
    for (int off = 16; off > 0; off >>= 1) {
        qs += __shfl_xor(qs, off, 32);
        ks += __shfl_xor(ks, off, 32);
    }
    if (lane == 0) { qn[p] = qs / (float)CO; kn[p] = ks / (float)CO; }
}

__device__ __forceinline__ float siluf(float v) { return v / (1.0f + expf(-v)); }

// ---------------------------------------------------------------------------
// Kernel 4: fused edge features -> MLP -> softmax -> message -> VN layernorm
// -> clamp. One block (128 threads = 1 thread/channel, 4 waves) per point.
// ---------------------------------------------------------------------------
__global__ __launch_bounds__(128) void attn_kernel(
    const float* __restrict__ x,  const float* __restrict__ Q,
    const float* __restrict__ Kt, const float* __restrict__ U,
    const float* __restrict__ qn, const float* __restrict__ kn,
    const int* __restrict__ idx,
    const float* __restrict__ W1, const float* __restrict__ b1,
    const float* __restrict__ W2, const float* __restrict__ b2,
    const float* __restrict__ W3, const float* __restrict__ b3,
    const float* __restrict__ gamma, const float* __restrict__ beta,
    float* __restrict__ out) {
    __shared__ float sQ[CO * 3];
    __shared__ float red[128];
    __shared__ float sDot[KK], sDist[KK], sKn[KK], sAttn[KK];
    __shared__ int   sNbr[KK];
    __shared__ float sW1[HH * 4], sB1[HH], sW2[HH * HH], sB2[HH], sW3[HH];

    const int t = threadIdx.x;
    const int p = blockIdx.x;
    const int b = p / NN;

    // stage MLP weights in LDS
    sW1[t] = W1[t];                                   // HH*4 == 128
    for (int e = t; e < HH * HH; e += 128) sW2[e] = W2[e];
    if (t < HH) { sB1[t] = b1[t]; sB2[t] = b2[t]; sW3[t] = W3[t]; }

    // this thread's channel of Q
    const float q0 = Q[p * (CO * 3) + t * 3 + 0];
    const float q1 = Q[p * (CO * 3) + t * 3 + 1];
    const float q2 = Q[p * (CO * 3) + t * 3 + 2];
    sQ[t * 3 + 0] = q0; sQ[t * 3 + 1] = q1; sQ[t * 3 + 2] = q2;

    if (t < KK) {
        const int g = b * NN + idx[p * KK + t];
        sNbr[t] = g;
        const float dx = x[p * 3 + 0] - x[g * 3 + 0];
        const float dy = x[p * 3 + 1] - x[g * 3 + 1];
        const float dz = x[p * 3 + 2] - x[g * 3 + 2];
        sDist[t] = sqrtf(dx * dx + dy * dy + dz * dz);
        sKn[t]   = kn[g];
    }
    __syncthreads();

    // mean_o sum_d Q[o,d]*K_nbr[k,o,d] : 4 waves x 4 edges each
    const int wave = t >> 5, lane = t & 31;
    for (int kk = 0; kk < 4; ++kk) {
        const int k = wave * 4 + kk;
        const float* kp = Kt + sNbr[k] * (CO * 3);
        float s = 0.f;
#pragma unroll
        for (int i = 0; i < 4; ++i) {
            const int c = lane + i * 32;
            s += sQ[c * 3 + 0] * kp[c * 3 + 0]
               + sQ[c * 3 + 1] * kp[c * 3 + 1]
               + sQ[c * 3 + 2] * kp[c * 3 + 2];
        }
#pragma unroll
        for (int off = 16; off > 0; off >>= 1) s += __shfl_xor(s, off, 32);
        if (lane == 0) sDot[k] = s / (float)CO;
    }
    __syncthreads();

    // edge MLP (4 -> 32 -> 32 -> 1, SiLU) on threads 0..15
    if (t < KK) {
        const float s0 = qn[p], s1 = sKn[t], s2 = sDot[t], s3 = sDist[t];
        float h1[HH];
#pragma unroll
        for (int j = 0; j < HH; ++j)
            h1[j] = siluf(sW1[j * 4 + 0] * s0 + sW1[j * 4 + 1] * s1 +
                          sW1[j * 4 + 2] * s2 + sW1[j * 4 + 3] * s3 + sB1[j]);
        float h2[HH];
#pragma unroll
        for (int j = 0; j < HH; ++j) {
            float a = sB2[j];
#pragma unroll
            for (int i = 0; i < HH; ++i) a += h1[i] * sW2[j * HH + i];
            h2[j] = siluf(a);
        }
        float logit = b3[0];
#pragma unroll
        for (int i = 0; i < HH; ++i) logit += h2[i] * sW3[i];
        logit = fminf(fmaxf(logit, -10.0f), 10.0f);
        sAttn[t] = logit;
    }
    __syncthreads();

    if (t == 0) {   // softmax over k=16
        float mx = sAttn[0];
        for (int k = 1; k < KK; ++k) mx = fmaxf(mx, sAttn[k]);
        float sum = 0.f;
        for (int k = 0; k < KK; ++k) { sAttn[k] = expf(sAttn[k] - mx); sum += sAttn[k]; }
        const float inv = 1.0f / sum;
        for (int k = 0; k < KK; ++k) sAttn[k] *= inv;
    }
    __syncthreads();

    // message aggregation + residual
    float m0 = 0.f, m1 = 0.f, m2 = 0.f;
#pragma unroll
    for (int k = 0; k < KK; ++k) {
        const float a = sAttn[k];
        const float* up = U + sNbr[k] * (CO * 3) + t * 3;
        m0 += a * up[0]; m1 += a * up[1]; m2 += a * up[2];
    }
    float o0 = q0 + 0.5f * m0, o1 = q1 + 0.5f * m1, o2 = q2 + 0.5f * m2;

    // VN layer norm over channels
    const float nrm = fmaxf(sqrtf(o0 * o0 + o1 * o1 + o2 * o2), EPSF);
    red[t] = nrm; __syncthreads();
    for (int s = 64; s > 0; s >>= 1) { if (t < s) red[t] += red[t + s]; __syncthreads(); }
    const float mean = red[0] / (float)CO; __syncthreads();
    const float dev = nrm - mean;
    red[t] = dev * dev; __syncthreads();
    for (int s = 64; s > 0; s >>= 1) { if (t < s) red[t] += red[t + s]; __syncthreads(); }
    const float stdv = fmaxf(sqrtf(red[0] / (float)(CO - 1)), EPSF);

    const float ns = (nrm - mean) / stdv * gamma[t] + beta[t];
    const float sc = fmaxf(ns, EPSF) / nrm;
    o0 *= sc; o1 *= sc; o2 *= sc;

    // clamp features
    const float cn = fmaxf(sqrtf(o0 * o0 + o1 * o1 + o2 * o2), EPSF);
    const float cs = fminf(CLAMP_MAXF / cn, 1.0f);
    out[p * (CO * 3) + t * 3 + 0] = o0 * cs;
    out[p * (CO * 3) + t * 3 + 1] = o1 * cs;
    out[p * (CO * 3) + t * 3 + 2] = o2 * cs;
}

// ---------------------------------------------------------------------------
extern "C" void kernel_launch(void* const* d_in, const int* in_sizes, int n_in,
                              void* d_out, int out_size, void* d_ws, size_t ws_size,
                              hipStream_t stream) {
    const float* x     = (const float*)d_in[0];
    const float* v     = (const float*)d_in[1];
    const float* Wq    = (const float*)d_in[2];
    const float* Wk    = (const float*)d_in[3];
    const float* Wu    = (const float*)d_in[4];
    const float* W1    = (const float*)d_in[5];
    const float* b1    = (const float*)d_in[6];
    const float* W2    = (const float*)d_in[7];
    const float* b2    = (const float*)d_in[8];
    const float* W3    = (const float*)d_in[9];
    const float* b3    = (const float*)d_in[10];
    const float* gamma = (const float*)d_in[11];
    const float* beta  = (const float*)d_in[12];

    // workspace layout: Q | Kt | U | qn | kn | idx   (~77 MB)
    float* ws = (float*)d_ws;
    const size_t feat = (size_t)BB * NN * CO * 3;
    float* Q  = ws;
    float* Kt = Q  + feat;
    float* U  = Kt + feat;
    float* qn = U  + feat;
    float* kn = qn + (size_t)BB * NN;
    int*   idx = (int*)(kn + (size_t)BB * NN);
    float* out = (float*)d_out;

    knn_kernel<<<dim3(BB * NN / 256), 256, 0, stream>>>(x, idx);
    vn_linear_kernel<<<dim3(BB * NN * 3 / 16, 2, 3), 128, 0, stream>>>(Wq, Wk, Wu, v, Q, Kt, U);
    norms_kernel<<<dim3(BB * NN / 8), 256, 0, stream>>>(Q, Kt, qn, kn);
    attn_kernel<<<dim3(BB * NN), 128, 0, stream>>>(x, Q, Kt, U, qn, kn, idx,
                                                   W1, b1, W2, b2, W3, b3,
                                                   gamma, beta, out);
}